// WESUP_60919816127007
// MI455X (gfx1250) — compile-verified
//
#include <hip/hip_runtime.h>
#include <math.h>

#define N_PIX_C  (224*224)
#define C_FEAT_C 4224
#define N_SEG_C  128
#define D_HID_C  1024
#define D_LOW_C  32

typedef __attribute__((ext_vector_type(2))) float v2f;
typedef __attribute__((ext_vector_type(8))) float v8f;

// ---------------------------------------------------------------- zero init
__global__ void WESUP_zero_kernel(float* __restrict__ p, int n) {
    int i = blockIdx.x * blockDim.x + threadIdx.x;
    if (i < n) p[i] = 0.0f;
}

// ------------------------------------------------- segment-sum over pixfeat
// Bandwidth-bound (848MB ~ 36us at 23.3TB/s). Each block owns a 64-channel
// chunk x pixel stripe. Each lane loads float4 (global_load_b128, 512B/wave)
// and accumulates into a 128-seg x 64-ch LDS tile (ds_add_f32); merged with
// global_atomic_add_f32 into the L2-resident seg_sum at the end.
#define SS_CH      64
#define SS_STRIPES 32
__global__ void WESUP_segsum_kernel(const float* __restrict__ pixfeat,
                                    const int* __restrict__ sp,
                                    float* __restrict__ seg_sum) {
    __shared__ float acc[N_SEG_C * SS_CH];   // 32 KB
    const int tid = threadIdx.x;             // 256 threads = 8 waves
    const int ch4 = (tid & 15) * 4;          // 16 lanes x float4 = 64 channels
    const int rg  = tid >> 4;                // 16 rows in flight per iteration
    const int ch_base = blockIdx.y * SS_CH;

    for (int i = tid; i < N_SEG_C * SS_CH; i += 256) acc[i] = 0.0f;
    __syncthreads();

    const int rows_per = (N_PIX_C + SS_STRIPES - 1) / SS_STRIPES;
    const int row_base = blockIdx.x * rows_per;
    const int row_end  = min(row_base + rows_per, N_PIX_C);

    for (int r = row_base + rg; r < row_end; r += 16) {
        const int s = sp[r];                 // broadcast across 16 lanes
        const float4 v = *(const float4*)(pixfeat + (size_t)r * C_FEAT_C
                                          + ch_base + ch4);   // b128, 16B-aligned
        float* dst = &acc[s * SS_CH + ch4];
        atomicAdd(&dst[0], v.x);             // ds_add_f32 x4
        atomicAdd(&dst[1], v.y);
        atomicAdd(&dst[2], v.z);
        atomicAdd(&dst[3], v.w);
    }
    __syncthreads();

    for (int i = tid; i < N_SEG_C * SS_CH; i += 256) {
        int s = i >> 6;
        int c = i & (SS_CH - 1);
        atomicAdd(&seg_sum[(size_t)s * C_FEAT_C + ch_base + c], acc[i]); // global_atomic_add_f32 (L2)
    }
}

// --------------------------------------------------- per-segment counts
__global__ void WESUP_counts_kernel(const int* __restrict__ sp,
                                    const int* __restrict__ y,
                                    float* __restrict__ counts,
                                    float* __restrict__ ccnt) {
    int i = blockIdx.x * blockDim.x + threadIdx.x;
    if (i < N_PIX_C) {
        int s = sp[i];
        atomicAdd(&counts[s], 1.0f);
        atomicAdd(&ccnt[s * 3 + y[i]], 1.0f);
    }
}

// ------------------------------- feat = seg_sum / max(cnt,1); fn = feat/||feat||
__global__ void WESUP_featnorm_kernel(float* __restrict__ feat,   // in: seg_sum, out: feat
                                      const float* __restrict__ counts,
                                      float* __restrict__ fn) {
    const int s   = blockIdx.x;
    const int tid = threadIdx.x;   // 256
    const float inv = 1.0f / fmaxf(counts[s], 1.0f);
    float ss = 0.0f;
    for (int c = tid; c < C_FEAT_C; c += 256) {
        float v = feat[(size_t)s * C_FEAT_C + c] * inv;
        feat[(size_t)s * C_FEAT_C + c] = v;
        ss += v * v;
    }
    __shared__ float red[256];
    red[tid] = ss;
    __syncthreads();
    for (int o = 128; o > 0; o >>= 1) {
        if (tid < o) red[tid] += red[tid + o];
        __syncthreads();
    }
    const float rn = 1.0f / sqrtf(red[0]);
    for (int c = tid; c < C_FEAT_C; c += 256)
        fn[(size_t)s * C_FEAT_C + c] = feat[(size_t)s * C_FEAT_C + c] * rn;
}

// --------------------------------------------------- per-segment label vote
__global__ void WESUP_label_kernel(const float* __restrict__ counts,
                                   const float* __restrict__ ccnt,
                                   int* __restrict__ lab) {
    int s = threadIdx.x;
    if (s < N_SEG_C) {
        float c0 = ccnt[s * 3 + 0], c1 = ccnt[s * 3 + 1], c2 = ccnt[s * 3 + 2];
        int best = 0; float bv = c0;
        if (c1 > bv) { bv = c1; best = 1; }
        if (c2 > bv) { bv = c2; best = 2; }
        int l = best;
        if (l == 0) {
            float cnt = counts[s];
            l = (c2 > floorf(cnt * 0.5f)) ? 2 : ((c1 >= 1.0f) ? 1 : 0);
        }
        lab[s] = l;
    }
}

// ------------------------------------------- aff = fn @ fn^T via fp32 WMMA
// One wave per 16x16 tile; K-loop in steps of 4 (V_WMMA_F32_16X16X4_F32).
__global__ void WESUP_affinity_kernel(const float* __restrict__ fn,
                                      float* __restrict__ aff) {
    const int lane = threadIdx.x;        // 32 = one wave, EXEC all ones
    const int hh   = lane >> 4;
    const int lid  = lane & 15;
    const int m0   = blockIdx.y * 16;
    const int n0   = blockIdx.x * 16;
    const float* arow = fn + (size_t)(m0 + lid) * C_FEAT_C;
    const float* brow = fn + (size_t)(n0 + lid) * C_FEAT_C;   // B = fn^T
    v8f acc = {};
    for (int k = 0; k < C_FEAT_C; k += 4) {
        const int ko = k + 2 * hh;
        v2f a, b;
        a.x = arow[ko]; a.y = arow[ko + 1];
        b.x = brow[ko]; b.y = brow[ko + 1];
        acc = __builtin_amdgcn_wmma_f32_16x16x4_f32(
            false, a, false, b, (short)0, acc, false, false);
    }
#pragma unroll
    for (int v = 0; v < 8; ++v)
        aff[(m0 + v + 8 * hh) * N_SEG_C + n0 + lid] = acc[v];
}

// --------------------------- C = [relu](A @ W + bias), fp32 WMMA 16x16x4
__global__ void WESUP_gemm_kernel(const float* __restrict__ A,
                                  const float* __restrict__ W,
                                  const float* __restrict__ bias,
                                  float* __restrict__ C,
                                  int M, int N, int K, int do_relu) {
    const int lane = threadIdx.x;        // 32 = one wave
    const int hh   = lane >> 4;
    const int lid  = lane & 15;
    const int m0   = blockIdx.y * 16;
    const int n0   = blockIdx.x * 16;
    const float* arow = A + (size_t)(m0 + lid) * K;
    v8f acc = {};
    for (int k = 0; k < K; k += 4) {
        const int ko = k + 2 * hh;
        v2f a, b;
        a.x = arow[ko];
        a.y = arow[ko + 1];
        b.x = W[(size_t)ko * N + n0 + lid];
        b.y = W[(size_t)(ko + 1) * N + n0 + lid];
        acc = __builtin_amdgcn_wmma_f32_16x16x4_f32(
            false, a, false, b, (short)0, acc, false, false);
    }
    const float bv = bias[n0 + lid];
#pragma unroll
    for (int v = 0; v < 8; ++v) {
        float val = acc[v] + bv;
        if (do_relu) val = fmaxf(val, 0.0f);
        C[(size_t)(m0 + v + 8 * hh) * N + n0 + lid] = val;
    }
}

// ------------------ label propagation + classifier head + softmax + outputs
__global__ void WESUP_finalize_kernel(const float* __restrict__ aff,
                                      const int* __restrict__ lab_pre,
                                      const float* __restrict__ h3,
                                      const float* __restrict__ wc,
                                      const float* __restrict__ bc,
                                      float* __restrict__ out) {
    int r = threadIdx.x;   // 128
    if (r < N_SEG_C) {
        int lp = lab_pre[r];
        float maxv = -INFINITY; int best = 0;
        for (int c = 0; c < N_SEG_C; ++c) {
            if (lab_pre[c] != 0) {
                float v = aff[r * N_SEG_C + c];
                if (v > maxv) { maxv = v; best = c; }   // first-max, as jnp.argmax
            }
        }
        int lf = (lp == 0 && maxv > 0.8f) ? lab_pre[best] : lp;

        float l0 = bc[0], l1 = bc[1];
        for (int j = 0; j < D_LOW_C; ++j) {
            float h = h3[r * D_LOW_C + j];
            l0 += h * wc[j * 2 + 0];
            l1 += h * wc[j * 2 + 1];
        }
        float mx = fmaxf(l0, l1);
        float e0 = expf(l0 - mx), e1 = expf(l1 - mx);
        float d  = e0 + e1;
        out[r * 2 + 0] = e0 / d;
        out[r * 2 + 1] = e1 / d;
        out[2 * N_SEG_C + r] = (float)lf;   // lab, concatenated after probs
    }
}

extern "C" void kernel_launch(void* const* d_in, const int* in_sizes, int n_in,
                              void* d_out, int out_size, void* d_ws, size_t ws_size,
                              hipStream_t stream) {
    const float* pixfeat = (const float*)d_in[0];
    const float* w1      = (const float*)d_in[1];
    const float* b1      = (const float*)d_in[2];
    const float* w2      = (const float*)d_in[3];
    const float* b2      = (const float*)d_in[4];
    const float* w3      = (const float*)d_in[5];
    const float* b3      = (const float*)d_in[6];
    const float* wc      = (const float*)d_in[7];
    const float* bc      = (const float*)d_in[8];
    const int*   sp      = (const int*)d_in[9];
    const int*   y       = (const int*)d_in[10];
    float* out = (float*)d_out;

    // workspace layout (floats)
    float* ws       = (float*)d_ws;
    float* seg_sum  = ws;                                   // 128*4224 (becomes feat)
    float* counts   = seg_sum + (size_t)N_SEG_C * C_FEAT_C; // 128
    float* ccnt     = counts + N_SEG_C;                     // 128*3
    float* fn       = ccnt + N_SEG_C * 3;                   // 128*4224
    float* aff      = fn + (size_t)N_SEG_C * C_FEAT_C;      // 128*128
    float* h1       = aff + N_SEG_C * N_SEG_C;              // 128*1024
    float* h2       = h1 + N_SEG_C * D_HID_C;               // 128*1024
    float* h3       = h2 + N_SEG_C * D_HID_C;               // 128*32
    int*   lab_pre  = (int*)(h3 + N_SEG_C * D_LOW_C);       // 128

    // 1) zero accumulators (every call: atomics must start from 0)
    const int nzero = N_SEG_C * C_FEAT_C + N_SEG_C + N_SEG_C * 3;
    WESUP_zero_kernel<<<(nzero + 255) / 256, 256, 0, stream>>>(seg_sum, nzero);

    // 2) segment sums (the 848MB bandwidth-bound phase, float4 loads)
    dim3 ssg(SS_STRIPES, C_FEAT_C / SS_CH);                 // 32 x 66
    WESUP_segsum_kernel<<<ssg, 256, 0, stream>>>(pixfeat, sp, seg_sum);

    // 3) counts + class counts
    WESUP_counts_kernel<<<(N_PIX_C + 255) / 256, 256, 0, stream>>>(sp, y, counts, ccnt);

    // 4) mean feature + L2-normalized copy
    WESUP_featnorm_kernel<<<N_SEG_C, 256, 0, stream>>>(seg_sum, counts, fn);

    // 5) per-segment label vote
    WESUP_label_kernel<<<1, N_SEG_C, 0, stream>>>(counts, ccnt, lab_pre);

    // 6) affinity = fn @ fn^T   (fp32 WMMA, 8x8 tiles of 16x16)
    WESUP_affinity_kernel<<<dim3(N_SEG_C / 16, N_SEG_C / 16), 32, 0, stream>>>(fn, aff);

    // 7) MLP: all fp32 WMMA, one wave per 16x16 tile
    WESUP_gemm_kernel<<<dim3(D_HID_C / 16, N_SEG_C / 16), 32, 0, stream>>>(
        seg_sum, w1, b1, h1, N_SEG_C, D_HID_C, C_FEAT_C, 1);
    WESUP_gemm_kernel<<<dim3(D_HID_C / 16, N_SEG_C / 16), 32, 0, stream>>>(
        h1, w2, b2, h2, N_SEG_C, D_HID_C, D_HID_C, 1);
    WESUP_gemm_kernel<<<dim3(D_LOW_C / 16, N_SEG_C / 16), 32, 0, stream>>>(
        h2, w3, b3, h3, N_SEG_C, D_LOW_C, D_HID_C, 1);

    // 8) label propagation + head + softmax -> d_out (probs | lab)
    WESUP_finalize_kernel<<<1, N_SEG_C, 0, stream>>>(aff, lab_pre, h3, wc, bc, out);
}